// PartialCRF_59167469470118
// MI455X (gfx1250) — compile-verified
//
#include <hip/hip_runtime.h>

#define T_LEN 1024
#define BN 32
#define DD 768
#define LL 48
#define NEGV (-1e20f)

typedef float v2f __attribute__((ext_vector_type(2)));
typedef float v8f __attribute__((ext_vector_type(8)));

// ---------------------------------------------------------------------------
// Kernel 1: emissions[t,b,l] = feats[t,b,:] @ W[:,l] + bias[l]
// Full-precision fp32 WMMA (V_WMMA_F32_16X16X4_F32). One wave per 16-row
// M-tile; N=48 handled as three 16-wide accumulators.
// ---------------------------------------------------------------------------
__global__ __launch_bounds__(256) void pcrf_emis_kernel(
    const float* __restrict__ feats, const float* __restrict__ W,
    const float* __restrict__ bias, float* __restrict__ em)
{
    const int warp  = threadIdx.x >> 5;
    const int lane  = threadIdx.x & 31;
    const int tile  = blockIdx.x * 8 + warp;        // 2048 tiles of 16 rows
    const int row0  = tile * 16;
    const int mrow  = lane & 15;
    const int khalf = (lane >> 4) * 2;              // 0 or 2
    const int n     = lane & 15;

    const float* arow = feats + (size_t)(row0 + mrow) * DD + khalf;
    const float* W0 = W + n;
    const float* W1 = W + n + 16;
    const float* W2 = W + n + 32;

    v8f c0 = {}, c1 = {}, c2 = {};

    #pragma unroll 4
    for (int kb = 0; kb < DD / 4; ++kb) {
        // A operand: lane&15 = M, lane>>4 selects K pair {0,1}/{2,3}
        v2f a = *(const v2f*)(arow + kb * 4);       // 8B aligned
        const int k0 = kb * 4 + khalf;
        v2f b0, b1, b2;
        b0.x = W0[k0 * LL]; b0.y = W0[k0 * LL + LL];
        b1.x = W1[k0 * LL]; b1.y = W1[k0 * LL + LL];
        b2.x = W2[k0 * LL]; b2.y = W2[k0 * LL + LL];
        c0 = __builtin_amdgcn_wmma_f32_16x16x4_f32(false, a, false, b0, (short)0, c0, false, false);
        c1 = __builtin_amdgcn_wmma_f32_16x16x4_f32(false, a, false, b1, (short)0, c1, false, false);
        c2 = __builtin_amdgcn_wmma_f32_16x16x4_f32(false, a, false, b2, (short)0, c2, false, false);
    }

    const float bn0 = bias[n], bn1 = bias[n + 16], bn2 = bias[n + 32];
    const int mhigh = (lane >> 4) * 8;
    #pragma unroll
    for (int i = 0; i < 8; ++i) {
        const size_t r = (size_t)(row0 + mhigh + i) * LL;
        em[r + n]      = c0[i] + bn0;
        em[r + n + 16] = c1[i] + bn1;
        em[r + n + 32] = c2[i] + bn2;
    }
}

// ---------------------------------------------------------------------------
// Kernel 2: CRF forward recursion. One wave32 per (batch, {fwd|partial}).
// Lane owns states j1=lane and j2=lane+32 (j2 valid for lanes 0..15).
// E = exp(trans_eff) precomputed in VGPRs; per step only 96 transcendentals.
// ---------------------------------------------------------------------------
__global__ __launch_bounds__(32) void pcrf_forward_kernel(
    const float* __restrict__ em, const unsigned char* __restrict__ tags,
    const int* __restrict__ lens, const float* __restrict__ beg,
    const float* __restrict__ trans, const float* __restrict__ endt,
    const unsigned char* __restrict__ bc, const unsigned char* __restrict__ ec,
    const unsigned char* __restrict__ tc, float* __restrict__ res)
{
    const int  b      = blockIdx.x;
    const bool constr = (blockIdx.y != 0);
    const int  lane   = threadIdx.x;
    const int  j1     = lane;
    const int  j2     = lane + 32;
    const bool has2   = (j2 < LL);
    const int  jj2    = has2 ? j2 : 0;              // clamped (safe loads)

    __shared__ __align__(16) float p[LL];

    // Precompute E rows (exp of effective transitions) into registers.
    float e1[LL], e2[LL];
    #pragma unroll
    for (int k = 0; k < LL; ++k) {
        const float t1 = trans[j1 * LL + k];
        const bool  m1 = constr && (tc[j1 * LL + k] != 0);
        e1[k] = m1 ? 0.f : __expf(t1);
        const float t2 = trans[jj2 * LL + k];
        const bool  m2 = constr && (tc[jj2 * LL + k] != 0);
        e2[k] = (!has2 || m2) ? 0.f : __expf(t2);
    }

    const float* emB = em + b * LL;                 // em[(t*BN + b)*LL + j]
    const unsigned char* tgB = tags + b * LL;
    const int STRD = BN * LL;                       // 1536

    // alpha_0
    float bg1 = beg[j1];  if (constr && bc[j1])  bg1 = NEGV;
    float a1  = emB[j1] + bg1;
    if (constr && tgB[j1]) a1 = NEGV;
    float bg2 = beg[jj2]; if (constr && bc[jj2]) bg2 = NEGV;
    float a2  = emB[jj2] + bg2;
    if (constr && tgB[jj2]) a2 = NEGV;
    if (!has2) a2 = NEGV;

    const int lenm1 = lens[b] - 1;                  // in [511, 1023]
    float la1 = a1, la2 = a2;

    // Software-pipelined emission/tag fetch, chunks of PF.
    constexpr int PF = 8;
    float cE1[PF], cE2[PF];
    int   cT1[PF], cT2[PF];
    #pragma unroll
    for (int u = 0; u < PF; ++u) {
        const int t = 1 + u;
        cE1[u] = emB[t * STRD + j1];
        cE2[u] = emB[t * STRD + jj2];
        cT1[u] = tgB[t * STRD + j1];
        cT2[u] = tgB[t * STRD + jj2];
    }

    for (int tb0 = 1; tb0 < T_LEN; tb0 += PF) {
        float nE1[PF], nE2[PF];
        int   nT1[PF], nT2[PF];
        #pragma unroll
        for (int u = 0; u < PF; ++u) {
            int tn = tb0 + PF + u;
            if (tn > T_LEN - 1) tn = T_LEN - 1;     // clamp: harmless refetch
            nE1[u] = emB[tn * STRD + j1];
            nE2[u] = emB[tn * STRD + jj2];
            nT1[u] = tgB[tn * STRD + j1];
            nT2[u] = tgB[tn * STRD + jj2];
        }
        #pragma unroll
        for (int u = 0; u < PF; ++u) {
            const int t = tb0 + u;
            if (t < T_LEN) {
                // global max of alpha across all 48 states
                float m = fmaxf(a1, a2);
                #pragma unroll
                for (int off = 16; off >= 1; off >>= 1)
                    m = fmaxf(m, __shfl_xor(m, off, 32));
                p[j1] = __expf(a1 - m);
                if (has2) p[j2] = __expf(a2 - m);
                __syncthreads();
                // s[j] = sum_k E[j][k] * p[k]   (pure fp32 matvec)
                float s1a = 0.f, s1b = 0.f, s2a = 0.f, s2b = 0.f;
                const float4* p4 = (const float4*)p;
                #pragma unroll
                for (int kq = 0; kq < LL / 4; ++kq) {
                    const float4 pv = p4[kq];
                    s1a = __builtin_fmaf(e1[4 * kq + 0], pv.x, s1a);
                    s1b = __builtin_fmaf(e1[4 * kq + 1], pv.y, s1b);
                    s1a = __builtin_fmaf(e1[4 * kq + 2], pv.z, s1a);
                    s1b = __builtin_fmaf(e1[4 * kq + 3], pv.w, s1b);
                    s2a = __builtin_fmaf(e2[4 * kq + 0], pv.x, s2a);
                    s2b = __builtin_fmaf(e2[4 * kq + 1], pv.y, s2b);
                    s2a = __builtin_fmaf(e2[4 * kq + 2], pv.z, s2a);
                    s2b = __builtin_fmaf(e2[4 * kq + 3], pv.w, s2b);
                }
                __syncthreads();
                float na1 = __logf(s1a + s1b) + m + cE1[u];
                float na2 = __logf(s2a + s2b) + m + cE2[u];
                if (constr && cT1[u]) na1 = NEGV;
                if (constr && cT2[u]) na2 = NEGV;
                a1 = na1;
                a2 = has2 ? na2 : NEGV;
                if (t == lenm1) { la1 = a1; la2 = a2; }
            }
        }
        #pragma unroll
        for (int u = 0; u < PF; ++u) {
            cE1[u] = nE1[u]; cE2[u] = nE2[u];
            cT1[u] = nT1[u]; cT2[u] = nT2[u];
        }
    }

    // logZ = logsumexp_j(alpha[len-1][j] + end_eff[j])
    float ev1 = endt[j1];  if (constr && ec[j1])  ev1 = NEGV;
    float ev2 = endt[jj2]; if (constr && ec[jj2]) ev2 = NEGV;
    float v1 = la1 + ev1;
    float v2 = has2 ? (la2 + ev2) : NEGV;
    float m2 = fmaxf(v1, v2);
    #pragma unroll
    for (int off = 16; off >= 1; off >>= 1)
        m2 = fmaxf(m2, __shfl_xor(m2, off, 32));
    float s = __expf(v1 - m2) + (has2 ? __expf(v2 - m2) : 0.f);
    #pragma unroll
    for (int off = 16; off >= 1; off >>= 1)
        s += __shfl_xor(s, off, 32);
    if (lane == 0) res[blockIdx.y * BN + b] = m2 + __logf(s);
}

// ---------------------------------------------------------------------------
// Kernel 3: out[b] = fwd[b] - par[b]
// ---------------------------------------------------------------------------
__global__ void pcrf_final_kernel(const float* __restrict__ res,
                                  float* __restrict__ out)
{
    const int b = threadIdx.x;
    if (b < BN) out[b] = res[b] - res[BN + b];
}

extern "C" void kernel_launch(void* const* d_in, const int* in_sizes, int n_in,
                              void* d_out, int out_size, void* d_ws, size_t ws_size,
                              hipStream_t stream) {
    (void)in_sizes; (void)n_in; (void)out_size; (void)ws_size;
    const float*         feats = (const float*)d_in[0];
    const unsigned char* tags  = (const unsigned char*)d_in[1];
    const int*           lens  = (const int*)d_in[2];
    const float*         W     = (const float*)d_in[3];
    const float*         bias  = (const float*)d_in[4];
    const float*         beg   = (const float*)d_in[5];
    const float*         trans = (const float*)d_in[6];
    const float*         endt  = (const float*)d_in[7];
    const unsigned char* bc    = (const unsigned char*)d_in[8];
    const unsigned char* ec    = (const unsigned char*)d_in[9];
    const unsigned char* tc    = (const unsigned char*)d_in[10];

    float* em  = (float*)d_ws;                          // T*B*L floats (6.3 MB)
    float* res = em + (size_t)T_LEN * BN * LL;          // 64 floats

    // emissions GEMM: 2048 M-tiles, 8 waves/block
    pcrf_emis_kernel<<<dim3(256), dim3(256), 0, stream>>>(feats, W, bias, em);
    // both recursions concurrently: grid (B, 2) -> y=0 fwd, y=1 partial
    pcrf_forward_kernel<<<dim3(BN, 2), dim3(32), 0, stream>>>(
        em, tags, lens, beg, trans, endt, bc, ec, tc, res);
    pcrf_final_kernel<<<dim3(1), dim3(32), 0, stream>>>(res, (float*)d_out);
}